// gc_sparse_32006096290305
// MI455X (gfx1250) — compile-verified
//
#include <hip/hip_runtime.h>

typedef __attribute__((ext_vector_type(2))) float v2f;
typedef __attribute__((ext_vector_type(8))) float v8f;

#define B_   16
#define C_   3
#define T_   256
#define V_   25
#define N_   48
#define MID_ 48
#define NB_  4
#define OUT_ 192
#define S_   6400      // T_*V_
#define KDIM 75        // V_*3
#define KPAD 76
#define HT_  128       // t-rows per half-panel
#define EPS_ 1e-5f

__device__ __forceinline__ float relu_(float x) { return x > 0.f ? x : 0.f; }

// ---------------------------------------------------------------------------
// Fused dilated-conv -> gate -> relu -> output projection -> squared loss.
// One workgroup per (v_w, n). 8 waves. Per (dilation, half): im2col panel
// As[128][76] + K-major weight panel wlsT[48][76] in LDS, then each wave does
// one 16x48 output tile via fully-unrolled f32 WMMA 16x16x4 (57 WMMAs/tile).
// ---------------------------------------------------------------------------
__global__ __launch_bounds__(256) void k_conv_fused(
    const float* __restrict__ x,    // (48,256,25)
    const float* __restrict__ Wb,   // (4,25,48,25,3)
    const float* __restrict__ bb,   // (4,25,48)
    const float* __restrict__ xg,   // (16,192) from ws
    const float* __restrict__ Ww,   // (25,192)
    const float* __restrict__ bw,   // (25)
    float* __restrict__ loss)       // (48,25,255)
{
    __shared__ float As[HT_][KPAD + 2];    // 128 x 78  (39,936 B)
    __shared__ float wlsT[MID_][80];       // 48 x 80   (15,360 B)
    __shared__ float bbs[MID_];
    __shared__ float xgs[OUT_];
    __shared__ float pf[T_];               // per-t accumulated projection

    const int vw   = blockIdx.x;
    const int n    = blockIdx.y;
    const int b    = n / C_;
    const int tid  = threadIdx.x;
    const int lane = tid & 31;
    const int wave = tid >> 5;
    const int half = lane >> 4;
    const int l16  = lane & 15;

    if (tid < OUT_) xgs[tid] = xg[b * OUT_ + tid];
    if (tid < T_)   pf[tid]  = 0.f;

    for (int i = 0; i < NB_; ++i) {
        const int d = i + 1;                       // DILS = (1,2,3,4)
        for (int h = 0; h < 2; ++h) {
            __syncthreads();                       // prior tiles done with LDS
            if (tid < HT_) {
                // build im2col panel rows t = h*128 + tid
                const int tg = h * HT_ + tid;
#pragma unroll
                for (int kk = 0; kk < KDIM; ++kk) {
                    const int vin = kk / 3, k = kk % 3;   // const-folded
                    const int tap = tg - (2 - k) * d;
                    As[tid][kk] = (tap >= 0) ? x[n * S_ + tap * V_ + vin] : 0.f;
                }
                As[tid][KDIM] = 0.f;               // K pad column
            } else if (h == 0) {
                // other 128 threads stage K-major weight panel (once per i)
                for (int idx = tid - HT_; idx < MID_ * KPAD; idx += HT_) {
                    const int m = idx / KPAD, kk = idx - m * KPAD;
                    float w = 0.f;
                    if (kk < KDIM) {
                        const int vin = kk / 3, k = kk % 3;
                        w = Wb[(((i * V_ + vw) * MID_ + m) * V_ + vin) * 3 + k];
                    }
                    wlsT[m][kk] = w;
                }
                const int mm = tid - HT_;
                if (mm < MID_) bbs[mm] = bb[(i * V_ + vw) * MID_ + mm];
                if (tid == HT_ && i + 1 < NB_)     // pre-pull next weights
                    __builtin_prefetch(&Wb[(((i + 1) * V_ + vw) * MID_) * V_ * 3], 0, 1);
            }
            __syncthreads();

            // one 16(t) x 48(m) tile per wave
            const int t0   = h * HT_ + wave * 16;  // global t base
            const int rloc = wave * 16 + l16;      // row within As
            const int cb   = half * 2;             // K sub-offset per lane
            v8f acc0 = {}; v8f acc1 = {}; v8f acc2 = {};
#pragma unroll
            for (int k0 = 0; k0 < KPAD; k0 += 4) {
                const v2f a  = *reinterpret_cast<const v2f*>(&As[rloc][cb + k0]);
                const v2f b0 = *reinterpret_cast<const v2f*>(&wlsT[ 0 + l16][cb + k0]);
                const v2f b1 = *reinterpret_cast<const v2f*>(&wlsT[16 + l16][cb + k0]);
                const v2f b2 = *reinterpret_cast<const v2f*>(&wlsT[32 + l16][cb + k0]);
                acc0 = __builtin_amdgcn_wmma_f32_16x16x4_f32(
                           false, a, false, b0, (short)0, acc0, false, false);
                acc1 = __builtin_amdgcn_wmma_f32_16x16x4_f32(
                           false, a, false, b1, (short)0, acc1, false, false);
                acc2 = __builtin_amdgcn_wmma_f32_16x16x4_f32(
                           false, a, false, b2, (short)0, acc2, false, false);
            }

            // Epilogue: gate by xg, relu, weight by Ww[v'], reduce 48 m's.
            // C/D layout: VGPR r, lanes 0-15 -> row t0+r, lanes 16-31 -> t0+r+8.
            const int ob = i * MID_;
#pragma unroll
            for (int r = 0; r < 8; ++r) {
                const int t  = t0 + r + half * 8;
                const int vp = (vw * T_ + t) % V_;
                const float* wwr = &Ww[vp * OUT_ + ob];
                float s;
                {
                    float g0 = acc0[r] + bbs[ 0 + l16];
                    float g1 = acc1[r] + bbs[16 + l16];
                    float g2 = acc2[r] + bbs[32 + l16];
                    s  = relu_(g0 * xgs[ob +  0 + l16]) * wwr[ 0 + l16];
                    s += relu_(g1 * xgs[ob + 16 + l16]) * wwr[16 + l16];
                    s += relu_(g2 * xgs[ob + 32 + l16]) * wwr[32 + l16];
                }
#pragma unroll
                for (int off = 8; off; off >>= 1) s += __shfl_xor(s, off, 16);
                if (l16 == 0) pf[t] += s;          // unique writer per t
            }
        }
    }
    __syncthreads();

    if (tid < T_) {
        const int t  = tid;
        const int jg = vw * T_ + t;
        const int tp = jg / V_, vp = jg % V_;
        if (tp < T_ - 1) {
            float pred = pf[t] + bw[vp];
            float tgt  = x[n * S_ + (tp + 1) * V_ + vp];
            float dd   = pred - tgt;
            loss[(n * V_ + vp) * (T_ - 1) + tp] = dd * dd;
        }
    }
}

// ---------------------------------------------------------------------------
// Gate path: small 1x1 convs + batchnorm (tiny fraction of FLOPs -> VALU)
// ---------------------------------------------------------------------------
__global__ void k_lin(const float* __restrict__ in, const float* __restrict__ w,
                      const float* __restrict__ bias, float* __restrict__ out,
                      int Bn, int Cin, int Cout)
{
    const int total = Bn * Cout * S_;
    for (int idx = blockIdx.x * blockDim.x + threadIdx.x; idx < total;
         idx += gridDim.x * blockDim.x) {
        int s = idx % S_;
        int o = (idx / S_) % Cout;
        int bq = idx / (S_ * Cout);
        float acc = bias[o];
        for (int c = 0; c < Cin; ++c)
            acc += in[(bq * Cin + c) * S_ + s] * w[o * Cin + c];
        out[idx] = acc;
    }
}

__global__ __launch_bounds__(256) void k_bnrelu(float* __restrict__ h,
    const float* __restrict__ g, const float* __restrict__ be, int Bn, int Cc)
{
    __shared__ float rs[256], rq[256];
    __shared__ float mv[2];
    const int o = blockIdx.x, tid = threadIdx.x;
    float s = 0.f, q = 0.f;
    for (int t = tid; t < Bn * S_; t += 256) {
        int bq = t / S_, ss = t % S_;
        float v = h[(bq * Cc + o) * S_ + ss];
        s += v; q += v * v;
    }
    rs[tid] = s; rq[tid] = q;
    __syncthreads();
    for (int off = 128; off; off >>= 1) {
        if (tid < off) { rs[tid] += rs[tid + off]; rq[tid] += rq[tid + off]; }
        __syncthreads();
    }
    if (tid == 0) {
        float inv = 1.f / (float)(Bn * S_);
        float m = rs[0] * inv;
        float var = rq[0] * inv - m * m;
        mv[0] = m; mv[1] = rsqrtf(var + EPS_);
    }
    __syncthreads();
    const float m = mv[0], rinv = mv[1], gg = g[o], bo = be[o];
    for (int t = tid; t < Bn * S_; t += 256) {
        int bq = t / S_, ss = t % S_;
        int id = (bq * Cc + o) * S_ + ss;
        float v = gg * (h[id] - m) * rinv + bo;
        h[id] = v > 0.f ? v : 0.f;
    }
}

__global__ __launch_bounds__(256) void k_head(
    const float* __restrict__ h3,                       // (16,1,6400)
    const float* __restrict__ cw0, const float* __restrict__ cb0,
    const float* __restrict__ cg0, const float* __restrict__ cbe0,
    const float* __restrict__ cw1, const float* __restrict__ cb1,
    const float* __restrict__ cg1, const float* __restrict__ cbe1,
    float* __restrict__ xg)                             // (16,192)
{
    __shared__ float hp[B_ * V_];      // [b*25+v]
    __shared__ float y0[100 * B_];     // [o*16+b]
    __shared__ float y1[OUT_ * B_];    // [o*16+b]
    const int tid = threadIdx.x;

    for (int idx = tid; idx < B_ * V_; idx += 256) {
        int bq = idx / V_, v = idx % V_;
        float s = 0.f;
        for (int t = 0; t < T_; ++t) s += h3[bq * S_ + t * V_ + v];
        hp[idx] = s * (1.f / T_);
    }
    __syncthreads();
    for (int idx = tid; idx < 100 * B_; idx += 256) {
        int o = idx / B_, bq = idx % B_;
        float a = cb0[o];
        for (int v = 0; v < V_; ++v) a += cw0[o * V_ + v] * hp[bq * V_ + v];
        y0[idx] = a;
    }
    __syncthreads();
    if (tid < 100) {
        int o = tid; float s = 0.f, q = 0.f;
        for (int bq = 0; bq < B_; ++bq) { float v = y0[o * B_ + bq]; s += v; q += v * v; }
        float m = s / B_, var = q / B_ - m * m, r = rsqrtf(var + EPS_);
        for (int bq = 0; bq < B_; ++bq) {
            float v = cg0[o] * (y0[o * B_ + bq] - m) * r + cbe0[o];
            y0[o * B_ + bq] = v > 0.f ? v : 0.f;
        }
    }
    __syncthreads();
    for (int idx = tid; idx < OUT_ * B_; idx += 256) {
        int o = idx / B_, bq = idx % B_;
        float a = cb1[o];
        for (int c = 0; c < 100; ++c) a += cw1[o * 100 + c] * y0[c * B_ + bq];
        y1[idx] = a;
    }
    __syncthreads();
    if (tid < OUT_) {
        int o = tid; float s = 0.f, q = 0.f;
        for (int bq = 0; bq < B_; ++bq) { float v = y1[o * B_ + bq]; s += v; q += v * v; }
        float m = s / B_, var = q / B_ - m * m, r = rsqrtf(var + EPS_);
        for (int bq = 0; bq < B_; ++bq) {
            float v = cg1[o] * (y1[o * B_ + bq] - m) * r + cbe1[o];
            xg[bq * OUT_ + o] = v > 0.f ? v : 0.f;
        }
    }
}

// gc / reg: one WG per (b,p); deterministic hierarchical reduction
__global__ __launch_bounds__(128) void k_gcreg(const float* __restrict__ Wb,
    const float* __restrict__ xg, float* __restrict__ gc, float* __restrict__ regpart)
{
    __shared__ float s2a[75], mxa[75];
    __shared__ float pr[128];
    const int blk = blockIdx.x;
    const int b = blk / V_, p = blk % V_;
    const int tid = threadIdx.x;
    if (tid < 75) {
        int q = tid / 3, k = tid % 3;
        float s2 = 0.f, mx = -3.4e38f;
        for (int o = 0; o < OUT_; ++o) {
            int i = o / MID_, m = o % MID_;
            float w = Wb[(((i * V_ + p) * MID_ + m) * V_ + q) * 3 + k];
            float t = w * xg[b * OUT_ + o];
            s2 += t * t;
            mx = fmaxf(mx, t);
        }
        s2a[tid] = s2; mxa[tid] = mx;
    }
    __syncthreads();
    float part = 0.f;
    if (tid < V_) {
        int q = tid;
        float a0 = s2a[q * 3], a1 = s2a[q * 3 + 1], a2 = s2a[q * 3 + 2];
        gc[(b * V_ + p) * V_ + q] =
            fmaxf(fmaxf(mxa[q * 3], mxa[q * 3 + 1]), mxa[q * 3 + 2]);
        part = sqrtf(a0 + a1 + a2) + sqrtf(a0) + sqrtf(a1) + sqrtf(a2);
    }
    pr[tid] = part;
    __syncthreads();
    for (int off = 64; off; off >>= 1) {
        if (tid < off) pr[tid] += pr[tid + off];
        __syncthreads();
    }
    if (tid == 0) regpart[blk] = pr[0];
}

__global__ __launch_bounds__(256) void k_regsum(const float* __restrict__ regpart,
                                                float* __restrict__ out)
{
    __shared__ float pr[256];
    const int tid = threadIdx.x;
    float s = 0.f;
    for (int i = tid; i < B_ * V_; i += 256) s += regpart[i];
    pr[tid] = s;
    __syncthreads();
    for (int off = 128; off; off >>= 1) {
        if (tid < off) pr[tid] += pr[tid + off];
        __syncthreads();
    }
    if (tid == 0) out[0] = 0.01f * pr[0];
}

// ---------------------------------------------------------------------------
extern "C" void kernel_launch(void* const* d_in, const int* in_sizes, int n_in,
                              void* d_out, int out_size, void* d_ws, size_t ws_size,
                              hipStream_t stream)
{
    const float* x  = (const float*)d_in[0];
    const float* Wb = (const float*)d_in[1];
    const float* bb = (const float*)d_in[2];
    const float* fw[4]  = {(const float*)d_in[3],  (const float*)d_in[7],
                           (const float*)d_in[11], (const float*)d_in[15]};
    const float* fb[4]  = {(const float*)d_in[4],  (const float*)d_in[8],
                           (const float*)d_in[12], (const float*)d_in[16]};
    const float* fg[4]  = {(const float*)d_in[5],  (const float*)d_in[9],
                           (const float*)d_in[13], (const float*)d_in[17]};
    const float* fbe[4] = {(const float*)d_in[6],  (const float*)d_in[10],
                           (const float*)d_in[14], (const float*)d_in[18]};
    const float* cw0 = (const float*)d_in[19]; const float* cb0 = (const float*)d_in[20];
    const float* cg0 = (const float*)d_in[21]; const float* cbe0= (const float*)d_in[22];
    const float* cw1 = (const float*)d_in[23]; const float* cb1 = (const float*)d_in[24];
    const float* cg1 = (const float*)d_in[25]; const float* cbe1= (const float*)d_in[26];
    const float* Ww  = (const float*)d_in[27]; const float* bw  = (const float*)d_in[28];

    float* out = (float*)d_out;
    float* ws  = (float*)d_ws;
    float* bufA    = ws;                         // 16*10*6400  = 1,024,000 f
    float* bufB    = ws + 1024000;               // 16*100*6400 = 10,240,000 f
    float* xg      = ws + 1024000 + 10240000;    // 3072 f
    float* regpart = xg + 4096;                  // 400 f

    const int dims[5] = {C_, 10, 100, 10, 1};
    float* bufs[2] = {bufA, bufB};
    const float* cur = x;
    for (int l = 0; l < 4; ++l) {
        float* dst = bufs[l & 1];
        int total = B_ * dims[l + 1] * S_;
        int blocks = (total + 255) / 256;
        k_lin<<<dim3(blocks), dim3(256), 0, stream>>>(cur, fw[l], fb[l], dst,
                                                      B_, dims[l], dims[l + 1]);
        k_bnrelu<<<dim3(dims[l + 1]), dim3(256), 0, stream>>>(dst, fg[l], fbe[l],
                                                              B_, dims[l + 1]);
        cur = dst;
    }
    // cur == bufB holds (16,1,6400)
    k_head<<<dim3(1), dim3(256), 0, stream>>>(cur, cw0, cb0, cg0, cbe0,
                                              cw1, cb1, cg1, cbe1, xg);
    k_gcreg<<<dim3(B_ * V_), dim3(128), 0, stream>>>(Wb, xg, out + 306000, regpart);
    k_regsum<<<dim3(1), dim3(256), 0, stream>>>(regpart, out + 316000);
    k_conv_fused<<<dim3(V_, N_), dim3(256), 0, stream>>>(x, Wb, bb, xg, Ww, bw, out);
}